// Wav2VecBertProcessor_74259984548145
// MI455X (gfx1250) — compile-verified
//
#include <hip/hip_runtime.h>
#include <hip/hip_bf16.h>

// ---------------------------------------------------------------------------
// Wav2Vec-BERT fbank front-end for MI455X (gfx1250), wave32 + WMMA f32.
//
// Roofline: ~200 MB HBM traffic -> ~10 us @ 23.3 TB/s. The binding term in
// round 1 was the 800 KB DFT basis re-streamed from L2 by 6016 blocks
// (4.8 GB). This version uses 64-frame blocks (1504 blocks) and register-
// reuses each B fragment across 4 M-tiles, so each basis stream is read once
// per block: L2 traffic 4.8 GB -> 1.2 GB, back near the HBM roofline.
//
//   k0: packed DFT cos/sin basis + packed mel filterbank
//   k1: 64-frame tile: raw window -> LDS, de-mean/preemph/window/mask -> LDS,
//       power spectrum via V_WMMA_F32_16X16X4_F32 (K=400, 16 WMMA/k-step),
//       mel projection via WMMA (K=256), log -> feats
//   k2: per-(b,mel) mean / unbiased var (8-way frame-split + LDS reduce)
//   k3: normalize, stack-by-2, pad row (1.0), attention mask
// ---------------------------------------------------------------------------

typedef float v2f __attribute__((ext_vector_type(2)));
typedef float v8f __attribute__((ext_vector_type(8)));

#define B_SZ        32
#define N_SAMP      480000
#define FRAME_LEN   400
#define HOP         160
#define NFRAMES     2998          // 1 + (480000-400)/160
#define FRAMES_BLK  64            // frames per block (4 M-tiles)
#define TILES64     47            // ceil(2998/64)
#define WIN_SAMP    (63 * HOP + FRAME_LEN)   // 10480 samples per block window
#define NMEL        80
#define MEL_FLOOR   1.192092955078125e-07f

// workspace layout (float offsets)
#define WS_COS   0                          // 100*256*2*2  = 102400 floats
#define WS_SIN   102400                     // 102400 floats
#define WS_MEL   204800                     // 64*80*2*2    = 20480 floats
#define WS_FEATS 225280                     // 32*2998*80   = 7674880 floats
#define WS_STATS (225280 + 7674880)         // 32*160 floats

// ---------------------------------------------------------------------------
// k0: packed tables.
// cos/sin pack: float2 index ((kg*256 + col)*2 + p) holds K rows {4kg+2p, +1}
//   -> a wave's B-fragment read (half=lane>>4 selects p, l=lane&15 selects
//      col) is 32 consecutive float2 = one 256B coalesced transaction.
// ---------------------------------------------------------------------------
__global__ void init_tables_k(const float* __restrict__ mel, float* __restrict__ ws) {
    int i = blockIdx.x * blockDim.x + threadIdx.x;
    float* cosP = ws + WS_COS;
    float* sinP = ws + WS_SIN;
    float* melP = ws + WS_MEL;
    if (i < 100 * 256) {
        int kg = i >> 8, c = i & 255;
        #pragma unroll
        for (int p = 0; p < 2; ++p) {
            #pragma unroll
            for (int q = 0; q < 2; ++q) {
                int n  = kg * 4 + 2 * p + q;          // time sample
                int ph = (n * c) & 511;               // exact mod-512 phase
                float th = (float)ph * (6.283185307179586f / 512.0f);
                cosP[(i * 2 + p) * 2 + q] = cosf(th);
                sinP[(i * 2 + p) * 2 + q] = sinf(th);
            }
        }
    }
    if (i < 64 * 80) {
        int kg = i / 80, c = i % 80;
        #pragma unroll
        for (int p = 0; p < 2; ++p)
            #pragma unroll
            for (int q = 0; q < 2; ++q)
                melP[(i * 2 + p) * 2 + q] = mel[(kg * 4 + 2 * p + q) * NMEL + c];
    }
}

// ---------------------------------------------------------------------------
// k1: fused frame-prep + DFT-GEMM + mel-GEMM for one 64-frame tile.
// 256 threads (8 waves). LDS: frW 103.4KB + shared(raw->power) 66.6KB = 170KB.
// ---------------------------------------------------------------------------
__global__ __launch_bounds__(256) void fbank_gemm_k(
    const float* __restrict__ wav, const float* __restrict__ maskp,
    const float* __restrict__ win,
    const float* __restrict__ cosP, const float* __restrict__ sinP,
    const float* __restrict__ melP, float* __restrict__ feats)
{
    __shared__ __align__(16) float frW[FRAMES_BLK * 404];  // stride 404: conflict-free
    __shared__ __align__(16) float shr[FRAMES_BLK * 260];  // rawS first, then power tile
    float* rawS = shr;                                     // 10480 floats
    float* pw   = shr;                                     // 64 x 260 after barrier

    const int b    = blockIdx.x / TILES64;
    const int tile = blockIdx.x % TILES64;
    const int f0   = tile * FRAMES_BLK;
    const int t    = threadIdx.x;
    const int lane = t & 31;
    const int wave = t >> 5;
    const int half = lane >> 4;      // WMMA fragment half
    const int l    = lane & 15;

    // ---- stage 0: raw sample window -> LDS (one coalesced pass) -----------
    const size_t sbase = (size_t)b * N_SAMP;
    for (int n = t; n < WIN_SAMP; n += 256) {
        int gi = min(f0 * HOP + n, N_SAMP - 1);            // clamp (last tile only)
        rawS[n] = wav[sbase + gi] * 32768.0f;
    }
    __syncthreads();

    // ---- stage 1: de-mean + pre-emphasis + window + mask (4 thr/frame) ----
    const int fi  = t >> 2;                                // 0..63
    const int sub = t & 3;
    const int fcl = min(f0 + fi, NFRAMES - 1);             // padded rows discarded later
    const int off = (fcl - f0) * HOP;
    float sum = 0.0f;
    for (int n = sub; n < FRAME_LEN; n += 4) sum += rawS[off + n];
    sum += __shfl_down(sum, 2, 4);
    sum += __shfl_down(sum, 1, 4);
    const float mean = __shfl(sum, lane & ~3) * (1.0f / FRAME_LEN);

    const float* msrc = maskp + sbase + (size_t)fcl * HOP;
    for (int n = sub; n < FRAME_LEN; n += 4) {
        float x = rawS[off + n];
        float y = (n == 0) ? (x - mean) * (1.0f - 0.97f)
                           : (x - 0.97f * rawS[off + n - 1] - (1.0f - 0.97f) * mean);
        frW[fi * 404 + n] = y * win[n] * msrc[n];
    }
    __syncthreads();   // frW ready; rawS reads done (pw may overwrite later)

    // ---- stage 2: power spectrum, DFT as f32 WMMA GEMM (K = 400) ----------
    // wave w owns bin tiles {w, w+8}; each B fragment register-reused across
    // the 4 M-tiles -> each basis stream read exactly once per block.
    v8f accC[2][4], accS[2][4];                            // 128 VGPRs
    #pragma unroll
    for (int p = 0; p < 2; ++p)
        #pragma unroll
        for (int mt = 0; mt < 4; ++mt) { accC[p][mt] = (v8f)0.0f; accS[p][mt] = (v8f)0.0f; }

    const v2f* cosT = (const v2f*)cosP;
    const v2f* sinT = (const v2f*)sinP;
    for (int kg = 0; kg < 100; ++kg) {
        const int kk = kg * 4;
        v2f bc[2], bs[2];
        #pragma unroll
        for (int p = 0; p < 2; ++p) {
            const int n0 = (wave + p * 8) * 16;
            const size_t bi = ((size_t)kg * 256 + n0 + l) * 2 + half;
            bc[p] = cosT[bi];
            bs[p] = sinT[bi];
        }
        #pragma unroll
        for (int mt = 0; mt < 4; ++mt) {
            // A 16x4 fragment: one aligned b64 LDS read per lane
            v2f a = *(const v2f*)&frW[(mt * 16 + l) * 404 + kk + 2 * half];
            #pragma unroll
            for (int p = 0; p < 2; ++p) {
                accC[p][mt] = __builtin_amdgcn_wmma_f32_16x16x4_f32(
                    false, a, false, bc[p], (short)0, accC[p][mt], false, false);
                accS[p][mt] = __builtin_amdgcn_wmma_f32_16x16x4_f32(
                    false, a, false, bs[p], (short)0, accS[p][mt], false, false);
            }
        }
    }
    __syncthreads();   // all waves done reading frW and rawS

    #pragma unroll
    for (int p = 0; p < 2; ++p) {
        const int n0 = (wave + p * 8) * 16;
        #pragma unroll
        for (int mt = 0; mt < 4; ++mt) {
            #pragma unroll
            for (int v = 0; v < 8; ++v) {
                const int m = mt * 16 + half * 8 + v;      // C/D layout: M = v + 8*half
                float c = accC[p][mt][v], s = accS[p][mt][v];
                pw[m * 260 + n0 + l] = c * c + s * s;      // stride 260: conflict-free
            }
        }
    }
    __syncthreads();

    // ---- stage 3: mel projection (K = 256) + log --------------------------
    if (wave < 5) {                                        // 5 mel N-tiles
        const int n0 = wave * 16;
        v8f acc[4];
        #pragma unroll
        for (int mt = 0; mt < 4; ++mt) acc[mt] = (v8f)0.0f;
        const v2f* melT = (const v2f*)melP;
        for (int kg = 0; kg < 64; ++kg) {
            const int kk = kg * 4;
            v2f bm = melT[((size_t)kg * 80 + n0 + l) * 2 + half];
            #pragma unroll
            for (int mt = 0; mt < 4; ++mt) {
                v2f a = *(const v2f*)&pw[(mt * 16 + l) * 260 + kk + 2 * half];
                acc[mt] = __builtin_amdgcn_wmma_f32_16x16x4_f32(
                    false, a, false, bm, (short)0, acc[mt], false, false);
            }
        }
        #pragma unroll
        for (int mt = 0; mt < 4; ++mt) {
            #pragma unroll
            for (int v = 0; v < 8; ++v) {
                const int fg = f0 + mt * 16 + half * 8 + v;
                if (fg < NFRAMES)
                    feats[((size_t)b * NFRAMES + fg) * NMEL + n0 + l] =
                        logf(fmaxf(acc[mt][v], MEL_FLOOR));
            }
        }
    }
}

// ---------------------------------------------------------------------------
// k2: per-(b, mel) mean and unbiased variance over frames.
// 640 threads: 8-way frame split per mel channel, LDS reduce.
// ---------------------------------------------------------------------------
__global__ __launch_bounds__(640) void stats_k(const float* __restrict__ feats,
                                               float* __restrict__ stats) {
    __shared__ float s1[8][NMEL];
    __shared__ float s2[8][NMEL];
    const int b  = blockIdx.x;
    const int t  = threadIdx.x;
    const int fp = t / NMEL;            // 0..7
    const int ml = t % NMEL;
    float s = 0.0f, q = 0.0f;
    for (int f = fp; f < NFRAMES; f += 8) {
        float v = feats[((size_t)b * NFRAMES + f) * NMEL + ml];
        s += v; q += v * v;
    }
    s1[fp][ml] = s; s2[fp][ml] = q;
    __syncthreads();
    if (fp == 0) {
        #pragma unroll
        for (int r = 1; r < 8; ++r) { s += s1[r][ml]; q += s2[r][ml]; }
        const float mean = s * (1.0f / NFRAMES);
        const float var  = (q - s * mean) * (1.0f / (NFRAMES - 1));
        stats[b * 160 + ml]        = mean;
        stats[b * 160 + NMEL + ml] = rsqrtf(var + 1e-7f);
    }
}

// ---------------------------------------------------------------------------
// k3: normalize, stack-by-2, pad frame (value 1.0), attention mask.
// out layout: [32*1500*160 floats][32*1500 int32]
// ---------------------------------------------------------------------------
__global__ __launch_bounds__(256) void finalize_k(const float* __restrict__ feats,
                                                  const float* __restrict__ stats,
                                                  float* __restrict__ out) {
    const size_t NFEAT = (size_t)B_SZ * 1500 * 160;
    const size_t i = (size_t)blockIdx.x * blockDim.x + threadIdx.x;
    if (i < NFEAT) {
        const int b   = (int)(i / (1500 * 160));
        const int r   = (int)(i % (1500 * 160));
        const int row = r / 160;
        const int j   = r % 160;
        float val;
        if (row < 1499) {
            const int hi  = (j >= NMEL);
            const int f   = row * 2 + hi;
            const int mel = j - hi * NMEL;
            val = (feats[((size_t)b * NFRAMES + f) * NMEL + mel] - stats[b * 160 + mel])
                  * stats[b * 160 + NMEL + mel];
        } else {
            val = 1.0f;                               // padding_value
        }
        out[i] = val;
    } else if (i < NFEAT + (size_t)B_SZ * 1500) {
        const size_t r = i - NFEAT;
        const int row = (int)(r % 1500);
        ((int*)out)[i] = (row < 1499) ? 1 : 0;
    }
}

// ---------------------------------------------------------------------------
extern "C" void kernel_launch(void* const* d_in, const int* in_sizes, int n_in,
                              void* d_out, int out_size, void* d_ws, size_t ws_size,
                              hipStream_t stream) {
    const float* wav   = (const float*)d_in[0];   // (32, 480000)
    const float* maskp = (const float*)d_in[1];   // (32, 480000)
    const float* mel   = (const float*)d_in[2];   // (257, 80)
    const float* win   = (const float*)d_in[3];   // (400,)
    float* ws  = (float*)d_ws;
    float* out = (float*)d_out;

    float* cosP  = ws + WS_COS;
    float* sinP  = ws + WS_SIN;
    float* melP  = ws + WS_MEL;
    float* feats = ws + WS_FEATS;
    float* stats = ws + WS_STATS;

    init_tables_k<<<(100 * 256 + 255) / 256, 256, 0, stream>>>(mel, ws);

    fbank_gemm_k<<<B_SZ * TILES64, 256, 0, stream>>>(
        wav, maskp, win, cosP, sinP, melP, feats);

    stats_k<<<B_SZ, 640, 0, stream>>>(feats, stats);

    const size_t total = (size_t)B_SZ * 1500 * 160 + (size_t)B_SZ * 1500;
    finalize_k<<<(unsigned)((total + 255) / 256), 256, 0, stream>>>(feats, stats, out);
}